// TactileNeuronAdapter_30683246362696
// MI455X (gfx1250) — compile-verified
//
#include <hip/hip_runtime.h>
#include <hip/hip_bf16.h>
#include <stdint.h>
#include <stddef.h>

// Izhikevich scan: I[B,T,N] -> (v_trace[B,T,N], spikes[B,T,N]) both f32.
// B=4, T=2000, N=4096. Parallelism = B*N = 16384 lanes = 512 wave32s.
// Per-wave deep latency hiding via TDM triple-buffered LDS staging.

#define B_DIM   4
#define T_DIM   2000
#define N_DIM   4096
#define TILE_N  32          // one wave's contiguous n-range (128 B rows)
#define CHUNK_T 40          // timesteps per TDM tile; 2000 = 40 * 50
#define NCHUNK  (T_DIM / CHUNK_T)
#define NBUF    3           // triple buffer: ~120 steps of prefetch cover

// Izhikevich constants (dt in ms)
#define DTMS   0.1f
#define P_A    0.02f
#define P_B    0.2f
#define P_C    (-65.0f)
#define P_D    8.0f
#define V_TH   30.0f

typedef uint32_t v4u __attribute__((ext_vector_type(4)));
typedef int      v8i __attribute__((ext_vector_type(8)));
typedef int      v4i __attribute__((ext_vector_type(4)));

#if defined(__gfx1250__) && __has_builtin(__builtin_amdgcn_tensor_load_to_lds)
#define HAVE_TDM 1
#else
#define HAVE_TDM 0
#endif

#if HAVE_TDM
__device__ __forceinline__ void tdm_issue_tile(const float* gsrc, uint32_t lds_off) {
    // D# group 0: count=1 | lds_addr | global_addr[56:0] | type=2
    uint64_t ga = (uint64_t)(uintptr_t)gsrc;
    v4u g0;
    g0.x = 1u;                                   // count=1, user descriptor, no gather
    g0.y = lds_off;                              // LDS byte address
    g0.z = (uint32_t)ga;                         // global_addr[31:0]
    g0.w = ((uint32_t)(ga >> 32) & 0x01FFFFFFu)  // global_addr[56:32]
         | (2u << 30);                           // type = 2 ("image")
    // D# group 1: mask=0 | data_size=4B | tdim0=32 | tdim1=40 | tile 32x40 | stride0=N
    v8i g1;
    g1[0] = (int)(2u << 16);                     // data_size = 2 -> 4 bytes
    g1[1] = (int)((uint32_t)TILE_N << 16);       // tensor_dim0[15:0] @ bits[63:48]
    g1[2] = (int)((uint32_t)CHUNK_T << 16);      // tensor_dim0 hi=0; tensor_dim1[15:0]
    g1[3] = (int)((uint32_t)TILE_N << 16);       // tensor_dim1 hi=0; tile_dim0
    g1[4] = (int)CHUNK_T;                        // tile_dim1; tile_dim2=0
    g1[5] = (int)N_DIM;                          // tensor_dim0_stride[31:0]
    g1[6] = 0;                                   // stride0 hi; stride1 lo
    g1[7] = 0;                                   // stride1 hi
    v4i g2 = {0, 0, 0, 0};
    v4i g3 = {0, 0, 0, 0};
#if __clang_major__ >= 23
    v8i g4 = {0, 0, 0, 0, 0, 0, 0, 0};
    __builtin_amdgcn_tensor_load_to_lds(g0, g1, g2, g3, g4, 0);
#else
    __builtin_amdgcn_tensor_load_to_lds(g0, g1, g2, g3, 0);
#endif
}
#endif

__device__ __forceinline__ void izh_step(float It, float& v, float& u,
                                         float& v_out, float& s_out) {
    float vv    = v * v;
    float acc   = __builtin_fmaf(0.04f, vv, 140.0f);
    acc         = __builtin_fmaf(5.0f, v, acc);
    acc        += (It - u);
    float v_new = __builtin_fmaf(DTMS, acc, v);
    float u_new = __builtin_fmaf(DTMS * P_A, __builtin_fmaf(P_B, v, -u), u);
    bool  spike = v_new >= V_TH;
    v_out = v_new;
    s_out = spike ? 1.0f : 0.0f;
    v     = spike ? P_C : v_new;
    u     = spike ? (u_new + P_D) : u_new;
}

__global__ __launch_bounds__(TILE_N) void izhikevich_scan_kernel(
    const float* __restrict__ in,
    float* __restrict__ vout,
    float* __restrict__ sout)
{
#if HAVE_TDM
    __shared__ float smem[NBUF * CHUNK_T * TILE_N];   // 15360 B, triple buffer
#endif
    const int lane = threadIdx.x;                  // 0..31
    const int wid  = blockIdx.x;                   // 0..511
    const int b    = wid >> 7;                     // N/TILE_N = 128 waves per batch
    const int n0   = (wid & 127) * TILE_N;
    const int n    = n0 + lane;

    const size_t base = (size_t)b * T_DIM * N_DIM + (size_t)n;
    float* vp = vout + base;
    float* sp = sout + base;

    float v = -65.0f;
    float u = P_B * -65.0f;

#if HAVE_TDM
    const float*   gsrc0 = in + (size_t)b * T_DIM * N_DIM + (size_t)n0;
    const uint32_t lds0  = (uint32_t)(uintptr_t)(&smem[0]);   // low 32 bits = LDS offset
    const uint32_t bufsz = (uint32_t)(CHUNK_T * TILE_N * sizeof(float));

    // Prime the pipeline: chunks 0..2 in flight.
    tdm_issue_tile(gsrc0,                              lds0);
    tdm_issue_tile(gsrc0 + (size_t)1 * CHUNK_T * N_DIM, lds0 + 1 * bufsz);
    tdm_issue_tile(gsrc0 + (size_t)2 * CHUNK_T * N_DIM, lds0 + 2 * bufsz);

    int bi = 0;  // rotating buffer index (0..NBUF-1)
    for (int c = 0; c < NCHUNK; ++c) {
        // Wait until chunk c has fully landed in LDS (TDM ops complete in order).
        const int rem = NCHUNK - 1 - c;   // chunks still to consume after this one
        if (rem >= 2)      __builtin_amdgcn_s_wait_tensorcnt(2);
        else if (rem == 1) __builtin_amdgcn_s_wait_tensorcnt(1);
        else               __builtin_amdgcn_s_wait_tensorcnt(0);
        __asm__ __volatile__("" ::: "memory");

        // Plain (non-volatile) addrspace(3) reads -> ds_load_b32 w/ imm offsets.
        const float* sm = smem + bi * (CHUNK_T * TILE_N) + lane;

        #pragma unroll
        for (int r = 0; r < CHUNK_T; ++r) {
            float It = sm[r * TILE_N];             // ds_load_b32, conflict-free
            float v_new, s;
            izh_step(It, v, u, v_new, s);
            __builtin_nontemporal_store(v_new, vp);  vp += N_DIM;
            __builtin_nontemporal_store(s,     sp);  sp += N_DIM;
        }

        // Refill the buffer we just drained with chunk c+NBUF.
        if (c + NBUF < NCHUNK) {
            __asm__ __volatile__("s_wait_dscnt 0" ::: "memory");
            tdm_issue_tile(gsrc0 + (size_t)(c + NBUF) * CHUNK_T * N_DIM,
                           lds0 + (uint32_t)bi * bufsz);
        }
        bi = (bi == NBUF - 1) ? 0 : bi + 1;
    }
#else
    // Fallback: direct NT loads with distant prefetch (no TDM builtin available).
    const float* ip = in + base;
    for (int t = 0; t < T_DIM; ++t) {
        if (t + 96 < T_DIM)
            __builtin_prefetch(ip + (size_t)(t + 96) * N_DIM, 0, 0);
        float It = __builtin_nontemporal_load(ip + (size_t)t * N_DIM);
        float v_new, s;
        izh_step(It, v, u, v_new, s);
        __builtin_nontemporal_store(v_new, vp);  vp += N_DIM;
        __builtin_nontemporal_store(s,     sp);  sp += N_DIM;
    }
#endif
}

extern "C" void kernel_launch(void* const* d_in, const int* in_sizes, int n_in,
                              void* d_out, int out_size, void* d_ws, size_t ws_size,
                              hipStream_t stream) {
    (void)in_sizes; (void)n_in; (void)out_size; (void)d_ws; (void)ws_size;
    const float* in   = (const float*)d_in[0];
    float*       vout = (float*)d_out;                                  // v_trace [B,T,N]
    float*       sout = vout + (size_t)B_DIM * T_DIM * N_DIM;           // spikes  [B,T,N]

    const int total_lanes = B_DIM * N_DIM;            // 16384
    dim3 grid(total_lanes / TILE_N);                  // 512 blocks
    dim3 block(TILE_N);                               // 1 wave32 per block
    izhikevich_scan_kernel<<<grid, block, 0, stream>>>(in, vout, sout);
}